// SE3EncoderDecoderQM9_87419764342920
// MI455X (gfx1250) — compile-verified
//
#include <hip/hip_runtime.h>

// SE(3)-Transformer (QM9) forward on MI455X / gfx1250.
// fp32 end-to-end; all dense GEMMs AND the equivariant message-passing use
// V_WMMA_F32_16X16X4_F32 (wave32 WMMA). ~3.2 GFLOP, <50MB working set ->
// L2-resident / latency bound, so we keep reference fp32 precision; the win
// is moving dot products onto the matrix pipe and amortizing pairwise loads
// across 16-wide output tiles.

typedef float v2f __attribute__((ext_vector_type(2)));
typedef float v8f __attribute__((ext_vector_type(8)));

constexpr int cB = 16, cN = 128, cC = 64, cH = 4, cDH = 16, cNR = 16, cIN = 11, cOUT = 19;
constexpr float cR = 10.0f, cEPS = 1e-8f;

static __device__ __forceinline__ v8f wmma4(v2f a, v2f b, v8f c) {
  // D = A(16x4) * B(4x16) + C(16x16), fp32
  return __builtin_amdgcn_wmma_f32_16x16x4_f32(false, a, false, b, (short)0, c, false, false);
}

// -------- fp32 WMMA GEMM, 16x64 strip per wave: dst = [relu](A[M,K]@W[K,N]+bias) + acc -----
// Requires N % 64 == 0, M % 16 == 0, K % 4 == 0 (true for every launch below).
__global__ void k_gemm(float* dst, const float* __restrict__ A, const float* __restrict__ W,
                       const float* __restrict__ bias, const float* acc,
                       int M, int K, int N, int relu) {
  int jt0 = blockIdx.x * 64, it = blockIdx.y * 16;
  int lane = threadIdx.x;
  int half = lane >> 4;      // 0: K pair {0,1}, 1: K pair {2,3}
  int sub  = lane & 15;      // A row / B col within tile
  int koff = half * 2;
  v8f c[4] = {};
  const float* Arow = A + (size_t)(it + sub) * K;
  for (int kk = 0; kk < K; kk += 4) {
    int k0 = kk + koff;
    v2f a; a.x = Arow[k0]; a.y = Arow[k0 + 1];
    const float* W0 = W + (size_t)k0 * N + jt0 + sub;
    const float* W1 = W + (size_t)(k0 + 1) * N + jt0 + sub;
#pragma unroll
    for (int t = 0; t < 4; ++t) {
      v2f b; b.x = W0[t * 16]; b.y = W1[t * 16];
      c[t] = wmma4(a, b, c[t]);
    }
  }
#pragma unroll
  for (int t = 0; t < 4; ++t) {
    int colg = jt0 + t * 16 + sub;
    float bb = bias ? bias[colg] : 0.0f;
    for (int r = 0; r < 8; ++r) {
      int row = it + half * 8 + r;           // C/D: VGPR r holds rows r and r+8
      float v = c[t][r] + bb;
      if (relu) v = fmaxf(v, 0.0f);
      if (acc)  v += acc[(size_t)row * N + colg];
      dst[(size_t)row * N + colg] = v;
    }
  }
}

// ---------------- fused RBF -> MLP(16->32 relu ->16) -> radial gates w[P,16] ----------------
__global__ void k_rbf_w(float* __restrict__ wout, const float* __restrict__ dist,
                        const float* __restrict__ rW1, const float* __restrict__ rb1,
                        const float* __restrict__ rW2, const float* __restrict__ rb2) {
  __shared__ float rh[16][33];
  int p0 = blockIdx.x * 16;
  int lane = threadIdx.x, half = lane >> 4, sub = lane & 15, koff = half * 2;
  float dd = dist[p0 + sub];
  const float inv = 1.0f / (2.0f * (cR / cNR) * (cR / cNR));
  v8f a0 = {}, a1 = {};
  for (int kk = 0; kk < 16; kk += 4) {
    int k0 = kk + koff;
    float c0 = cR * (float)k0 / 15.0f, c1 = cR * (float)(k0 + 1) / 15.0f;
    v2f a;  a.x = expf(-(dd - c0) * (dd - c0) * inv);
            a.y = expf(-(dd - c1) * (dd - c1) * inv);
    v2f b0; b0.x = rW1[k0 * 32 + sub];       b0.y = rW1[(k0 + 1) * 32 + sub];
    v2f b1; b1.x = rW1[k0 * 32 + sub + 16];  b1.y = rW1[(k0 + 1) * 32 + sub + 16];
    a0 = wmma4(a, b0, a0);
    a1 = wmma4(a, b1, a1);
  }
  for (int r = 0; r < 8; ++r) {
    int row = half * 8 + r;
    rh[row][sub]      = fmaxf(a0[r] + rb1[sub], 0.0f);
    rh[row][sub + 16] = fmaxf(a1[r] + rb1[sub + 16], 0.0f);
  }
  __syncthreads();
  v8f c = {};
  for (int kk = 0; kk < 32; kk += 4) {
    int k0 = kk + koff;
    v2f a; a.x = rh[sub][k0];          a.y = rh[sub][k0 + 1];
    v2f b; b.x = rW2[k0 * 16 + sub];   b.y = rW2[(k0 + 1) * 16 + sub];
    c = wmma4(a, b, c);
  }
  for (int r = 0; r < 8; ++r) {
    int row = p0 + half * 8 + r;
    wout[(size_t)row * 16 + sub] = c[r] + rb2[sub];
  }
}

// ---------------- attention logits: sim = (q0.k0 * w00 + q1.k1 * w11)*scale, masked ----------
__global__ void k_sim(float* __restrict__ sim, const float* __restrict__ q0,
                      const float* __restrict__ k0, const float* __restrict__ q1,
                      const float* __restrict__ k1, const float* __restrict__ w,
                      const float* __restrict__ nmf) {
  int j0 = blockIdx.x * 16, i0 = blockIdx.y * 16;
  int b = blockIdx.z / cH, h = blockIdx.z % cH;
  int lane = threadIdx.x, half = lane >> 4, sub = lane & 15, koff = half * 2;
  v8f s0 = {}, s1 = {};
  const float* q0r = q0 + ((size_t)(b * cN + i0 + sub) * cH + h) * cDH;
  const float* k0r = k0 + ((size_t)(b * cN + j0 + sub) * cH + h) * cDH;
  for (int kk = 0; kk < cDH; kk += 4) {
    int k0i = kk + koff;
    v2f a;  a.x = q0r[k0i];  a.y = q0r[k0i + 1];
    v2f bb; bb.x = k0r[k0i]; bb.y = k0r[k0i + 1];
    s0 = wmma4(a, bb, s0);
  }
  const float* q1r = q1 + (size_t)(b * cN + i0 + sub) * 3 * cC + h * cDH;
  const float* k1r = k1 + (size_t)(b * cN + j0 + sub) * 3 * cC + h * cDH;
  for (int kk = 0; kk < 48; kk += 4) {   // K dim = (v,d) = 3*16
    int kidx = kk + koff; int v = kidx >> 4, d = kidx & 15;
    v2f a;  a.x = q1r[v * cC + d];  a.y = q1r[v * cC + d + 1];
    v2f bb; bb.x = k1r[v * cC + d]; bb.y = k1r[v * cC + d + 1];
    s1 = wmma4(a, bb, s1);
  }
  const float scale = 0.25f;  // DH^-0.5
  for (int r = 0; r < 8; ++r) {
    int i = i0 + half * 8 + r, j = j0 + sub;
    size_t pij = (size_t)(b * cN + i) * cN + j;
    float w00 = w[pij * 16 + 0 * cH + h];
    float w11 = w[pij * 16 + 3 * cH + h];
    float val = (s0[r] * w00 + s1[r] * w11) * scale;
    sim[((size_t)(b * cH + h) * cN + i) * cN + j] = (nmf[pij] > 0.5f) ? val : -1e9f;
  }
}

// ---- WMMA message passing: one wave per (b,h,16-row strip); K-loop over neighbors j ----
// out0 = (a*w00)@v0 + sum_v (a*w10*Y1v)@v1v ; out1v = (a*w01*Y1v)@v0 + (a*w11)@v1v
__global__ void k_attn_out(float* __restrict__ out0, float* __restrict__ out1,
                           const float* __restrict__ attn, const float* __restrict__ w,
                           const float* __restrict__ Y1, const float* __restrict__ v0,
                           const float* __restrict__ v1) {
  int i0 = blockIdx.x * 16;
  int b = blockIdx.y / cH, h = blockIdx.y % cH;
  int lane = threadIdx.x, half = lane >> 4, sub = lane & 15, koff = half * 2;
  v8f c0 = {}, c1x = {}, c1y = {}, c1z = {};
  int i = i0 + sub;                                    // A-matrix row for this lane
  const float* arow = attn + ((size_t)(b * cH + h) * cN + i) * cN;
  const float* wrow = w  + (size_t)(b * cN + i) * cN * 16;
  const float* yrow = Y1 + (size_t)(b * cN + i) * cN * 3;
  for (int kk = 0; kk < cN; kk += 4) {
    int ja = kk + koff, jb = ja + 1;                   // this lane's two K indices
    // build the 8 A-matrix values (per K element) from gates a*w and Y1
    float aa = arow[ja], ab = arow[jb];
    float w00a = wrow[ja * 16 + h],            w00b = wrow[jb * 16 + h];
    float w01a = wrow[ja * 16 + cH + h],       w01b = wrow[jb * 16 + cH + h];
    float w10a = wrow[ja * 16 + 2 * cH + h],   w10b = wrow[jb * 16 + 2 * cH + h];
    float w11a = wrow[ja * 16 + 3 * cH + h],   w11b = wrow[jb * 16 + 3 * cH + h];
    float yxa = yrow[ja * 3], yya = yrow[ja * 3 + 1], yza = yrow[ja * 3 + 2];
    float yxb = yrow[jb * 3], yyb = yrow[jb * 3 + 1], yzb = yrow[jb * 3 + 2];
    v2f p00;  p00.x = aa * w00a;        p00.y = ab * w00b;
    v2f p11;  p11.x = aa * w11a;        p11.y = ab * w11b;
    float g10a = aa * w10a, g10b = ab * w10b;
    float g01a = aa * w01a, g01b = ab * w01b;
    v2f p10x; p10x.x = g10a * yxa; p10x.y = g10b * yxb;
    v2f p10y; p10y.x = g10a * yya; p10y.y = g10b * yyb;
    v2f p10z; p10z.x = g10a * yza; p10z.y = g10b * yzb;
    v2f p01x; p01x.x = g01a * yxa; p01x.y = g01b * yxb;
    v2f p01y; p01y.x = g01a * yya; p01y.y = g01b * yyb;
    v2f p01z; p01z.x = g01a * yza; p01z.y = g01b * yzb;
    // B tiles: rows j = ja, jb ; cols d = sub (DH == 16 -> exactly one tile)
    const float* v0a = v0 + (size_t)(b * cN + ja) * cC + h * cDH;
    const float* v0b = v0 + (size_t)(b * cN + jb) * cC + h * cDH;
    v2f bv0; bv0.x = v0a[sub]; bv0.y = v0b[sub];
    const float* v1a = v1 + (size_t)(b * cN + ja) * 3 * cC + h * cDH;
    const float* v1b = v1 + (size_t)(b * cN + jb) * 3 * cC + h * cDH;
    v2f bvx; bvx.x = v1a[sub];          bvx.y = v1b[sub];
    v2f bvy; bvy.x = v1a[cC + sub];     bvy.y = v1b[cC + sub];
    v2f bvz; bvz.x = v1a[2 * cC + sub]; bvz.y = v1b[2 * cC + sub];
    c0  = wmma4(p00,  bv0, c0);
    c0  = wmma4(p10x, bvx, c0);
    c0  = wmma4(p10y, bvy, c0);
    c0  = wmma4(p10z, bvz, c0);
    c1x = wmma4(p01x, bv0, c1x);
    c1x = wmma4(p11,  bvx, c1x);
    c1y = wmma4(p01y, bv0, c1y);
    c1y = wmma4(p11,  bvy, c1y);
    c1z = wmma4(p01z, bv0, c1z);
    c1z = wmma4(p11,  bvz, c1z);
  }
  for (int r = 0; r < 8; ++r) {
    int row = i0 + half * 8 + r;
    out0[(size_t)(b * cN + row) * cC + h * cDH + sub] = c0[r];
    size_t ob = (size_t)(b * cN + row) * 3 * cC + h * cDH + sub;
    out1[ob]          = c1x[r];
    out1[ob + cC]     = c1y[r];
    out1[ob + 2 * cC] = c1z[r];
  }
}

// ---------------- small VALU kernels ----------------
__global__ void k_embed(float* __restrict__ f0, const float* __restrict__ feats,
                        const float* __restrict__ W, const float* __restrict__ bias) {
  int t = blockIdx.x * blockDim.x + threadIdx.x;
  if (t >= cB * cN * cC) return;
  int c = t % cC, row = t / cC;
  float s = bias[c];
  for (int k = 0; k < cIN; ++k) s += feats[(size_t)row * cIN + k] * W[k * cC + c];
  f0[t] = s;
}

__global__ void k_zero(float* p, int n) {
  int t = blockIdx.x * blockDim.x + threadIdx.x;
  if (t < n) p[t] = 0.0f;
}

__global__ void k_pair(float* __restrict__ dist, float* __restrict__ Y1, float* __restrict__ nmf,
                       const float* __restrict__ coords, const unsigned char* __restrict__ mask,
                       const unsigned char* __restrict__ adj) {
  int t = blockIdx.x * blockDim.x + threadIdx.x;
  if (t >= cB * cN * cN) return;
  int j = t % cN, i = (t / cN) % cN, b = t / (cN * cN);
  const float* ci = coords + (size_t)(b * cN + i) * 3;
  const float* cj = coords + (size_t)(b * cN + j) * 3;
  float rx = cj[0] - ci[0], ry = cj[1] - ci[1], rz = cj[2] - ci[2];
  float d = sqrtf(rx * rx + ry * ry + rz * rz + cEPS);
  dist[t] = d;
  float id = 1.0f / d;
  Y1[(size_t)t * 3 + 0] = rx * id;
  Y1[(size_t)t * 3 + 1] = ry * id;
  Y1[(size_t)t * 3 + 2] = rz * id;
  bool nm = ((d <= cR) || adj[t]) && mask[b * cN + i] && mask[b * cN + j] && (i != j);
  nmf[t] = nm ? 1.0f : 0.0f;
}

__global__ void k_ln(float* __restrict__ dst, const float* __restrict__ src,
                     const float* __restrict__ g, const float* __restrict__ bb) {
  __shared__ float red[cC];
  int row = blockIdx.x, c = threadIdx.x;
  float x = src[(size_t)row * cC + c];
  red[c] = x; __syncthreads();
  for (int s = 32; s > 0; s >>= 1) { if (c < s) red[c] += red[c + s]; __syncthreads(); }
  float m = red[0] / cC; __syncthreads();
  float dx = x - m;
  red[c] = dx * dx; __syncthreads();
  for (int s = 32; s > 0; s >>= 1) { if (c < s) red[c] += red[c + s]; __syncthreads(); }
  float var = red[0] / cC;
  dst[(size_t)row * cC + c] = dx * rsqrtf(var + 1e-5f) * g[c] + bb[c];
}

__global__ void k_softmax(float* __restrict__ a, const float* __restrict__ nmf) {
  __shared__ float red[cN];
  int row = blockIdx.x;                 // row = (b*H+h)*N + i
  int j = threadIdx.x;
  int i = row % cN, bh = row / cN, b = bh / cH;
  float v = a[(size_t)row * cN + j];
  red[j] = v; __syncthreads();
  for (int s = 64; s > 0; s >>= 1) { if (j < s) red[j] = fmaxf(red[j], red[j + s]); __syncthreads(); }
  float mx = red[0]; __syncthreads();
  float e = expf(v - mx);
  red[j] = e; __syncthreads();
  for (int s = 64; s > 0; s >>= 1) { if (j < s) red[j] += red[j + s]; __syncthreads(); }
  float sum = red[0];
  a[(size_t)row * cN + j] = e / sum * nmf[(size_t)(b * cN + i) * cN + j];
}

__global__ void k_norm1(float* __restrict__ f1, const float* __restrict__ g,
                        const float* __restrict__ bb) {
  int t = blockIdx.x * blockDim.x + threadIdx.x;
  if (t >= cB * cN * cC) return;
  int c = t % cC, row = t / cC;
  size_t base = (size_t)row * 3 * cC + c;
  float x = f1[base], y = f1[base + cC], z = f1[base + 2 * cC];
  float n = sqrtf(x * x + y * y + z * z + cEPS);
  float gate = fmaxf(n * g[c] + bb[c], 0.0f) / n;
  f1[base] = x * gate; f1[base + cC] = y * gate; f1[base + 2 * cC] = z * gate;
}

__global__ void k_pool_out(float* __restrict__ out, const float* __restrict__ f0,
                           const unsigned char* __restrict__ mask,
                           const float* __restrict__ Wout, const float* __restrict__ bout) {
  __shared__ float pooled[cC];
  int b = blockIdx.x, c = threadIdx.x;
  float s = 0.f, cnt = 0.f;
  for (int n = 0; n < cN; ++n) {
    float m = mask[b * cN + n] ? 1.0f : 0.0f;
    s += f0[(size_t)(b * cN + n) * cC + c] * m;
    cnt += m;
  }
  pooled[c] = s / fmaxf(cnt, 1.0f);
  __syncthreads();
  if (c < cOUT) {
    float o = bout[c];
    for (int k = 0; k < cC; ++k) o += pooled[k] * Wout[k * cOUT + c];
    out[b * cOUT + c] = o;
  }
}

// ------------------------------------ host launcher ------------------------------------
extern "C" void kernel_launch(void* const* d_in, const int* in_sizes, int n_in,
                              void* d_out, int out_size, void* d_ws, size_t ws_size,
                              hipStream_t stream) {
  (void)in_sizes; (void)n_in; (void)out_size; (void)ws_size;
  const float* feats  = (const float*)d_in[0];
  const float* coords = (const float*)d_in[1];
  const float* W_emb  = (const float*)d_in[2];
  const float* b_emb  = (const float*)d_in[3];
  const float* Wq0 = (const float*)d_in[4];
  const float* Wk0 = (const float*)d_in[5];
  const float* Wv0 = (const float*)d_in[6];
  const float* Wq1 = (const float*)d_in[7];
  const float* Wk1 = (const float*)d_in[8];
  const float* Wv1 = (const float*)d_in[9];
  const float* rW1 = (const float*)d_in[10];
  const float* rb1 = (const float*)d_in[11];
  const float* rW2 = (const float*)d_in[12];
  const float* rb2 = (const float*)d_in[13];
  const float* Wo0 = (const float*)d_in[14];
  const float* bo0 = (const float*)d_in[15];
  const float* Wo1 = (const float*)d_in[16];
  const float* g0  = (const float*)d_in[17];
  const float* b0  = (const float*)d_in[18];
  const float* gn  = (const float*)d_in[19];
  const float* bn  = (const float*)d_in[20];
  const float* g2  = (const float*)d_in[21];
  const float* b2  = (const float*)d_in[22];
  const float* F1  = (const float*)d_in[23];
  const float* fb1 = (const float*)d_in[24];
  const float* F2  = (const float*)d_in[25];
  const float* fb2 = (const float*)d_in[26];
  const float* Wf1 = (const float*)d_in[27];
  const float* Wout = (const float*)d_in[28];
  const float* bout = (const float*)d_in[29];
  const unsigned char* mask = (const unsigned char*)d_in[30];
  const unsigned char* adj  = (const unsigned char*)d_in[31];

  float* ws = (float*)d_ws;
  size_t off = 0;
  auto carve = [&](size_t n) { float* p = ws + off; off += (n + 63) & ~(size_t)63; return p; };
  const int P = cB * cN * cN;                 // 262144 pairs
  float* dist = carve(P);
  float* Y1   = carve((size_t)P * 3);
  float* nmf  = carve(P);
  float* wbuf = carve((size_t)P * 16);
  float* sim  = carve((size_t)cB * cH * cN * cN);
  float* f0   = carve(cB * cN * cC);
  float* f1   = carve(cB * cN * 3 * cC);
  float* f1b  = carve(cB * cN * 3 * cC);
  float* x0   = carve(cB * cN * cC);
  float* q0   = carve(cB * cN * cC);
  float* k0b  = carve(cB * cN * cC);
  float* v0   = carve(cB * cN * cC);
  float* q1   = carve(cB * cN * 3 * cC);
  float* k1b  = carve(cB * cN * 3 * cC);
  float* v1   = carve(cB * cN * 3 * cC);
  float* o0   = carve(cB * cN * cC);
  float* o1   = carve(cB * cN * 3 * cC);
  float* ffnh = carve(cB * cN * 4 * cC);

  const int M0 = cB * cN, M1 = cB * cN * 3;
  k_embed<<<(M0 * cC + 255) / 256, 256, 0, stream>>>(f0, feats, W_emb, b_emb);
  k_zero<<<(M1 * cC + 255) / 256, 256, 0, stream>>>(f1, M1 * cC);
  k_pair<<<(P + 255) / 256, 256, 0, stream>>>(dist, Y1, nmf, coords, mask, adj);

  for (int l = 0; l < 2; ++l) {
    k_rbf_w<<<P / 16, 32, 0, stream>>>(wbuf, dist, rW1 + l * 16 * 32, rb1 + l * 32,
                                       rW2 + l * 32 * 16, rb2 + l * 16);
    k_ln<<<M0, cC, 0, stream>>>(x0, f0, g0 + l * cC, b0 + l * cC);

    dim3 g64(cC / 64, M0 / 16);          // (1, 128)
    k_gemm<<<g64, 32, 0, stream>>>(q0, x0, Wq0 + l * 4096, nullptr, nullptr, M0, cC, cC, 0);
    k_gemm<<<g64, 32, 0, stream>>>(k0b, x0, Wk0 + l * 4096, nullptr, nullptr, M0, cC, cC, 0);
    k_gemm<<<g64, 32, 0, stream>>>(v0, x0, Wv0 + l * 4096, nullptr, nullptr, M0, cC, cC, 0);
    dim3 g64b(cC / 64, M1 / 16);         // (1, 384)
    k_gemm<<<g64b, 32, 0, stream>>>(q1, f1, Wq1 + l * 4096, nullptr, nullptr, M1, cC, cC, 0);
    k_gemm<<<g64b, 32, 0, stream>>>(k1b, f1, Wk1 + l * 4096, nullptr, nullptr, M1, cC, cC, 0);
    k_gemm<<<g64b, 32, 0, stream>>>(v1, f1, Wv1 + l * 4096, nullptr, nullptr, M1, cC, cC, 0);

    dim3 gs(cN / 16, cN / 16, cB * cH);
    k_sim<<<gs, 32, 0, stream>>>(sim, q0, k0b, q1, k1b, wbuf, nmf);
    k_softmax<<<cB * cH * cN, cN, 0, stream>>>(sim, nmf);
    dim3 ga(cN / 16, cB * cH);
    k_attn_out<<<ga, 32, 0, stream>>>(o0, o1, sim, wbuf, Y1, v0, v1);

    k_gemm<<<g64, 32, 0, stream>>>(f0, o0, Wo0 + l * 4096, bo0 + l * cC, f0, M0, cC, cC, 0);
    k_gemm<<<g64b, 32, 0, stream>>>(f1, o1, Wo1 + l * 4096, nullptr, f1, M1, cC, cC, 0);
    k_norm1<<<(M0 * cC + 255) / 256, 256, 0, stream>>>(f1, gn + l * cC, bn + l * cC);

    k_ln<<<M0, cC, 0, stream>>>(x0, f0, g2 + l * cC, b2 + l * cC);
    dim3 gf1(4 * cC / 64, M0 / 16);      // (4, 128)
    k_gemm<<<gf1, 32, 0, stream>>>(ffnh, x0, F1 + l * cC * 4 * cC, fb1 + l * 4 * cC, nullptr,
                                   M0, cC, 4 * cC, 1);
    k_gemm<<<g64, 32, 0, stream>>>(f0, ffnh, F2 + l * 4 * cC * cC, fb2 + l * cC, f0,
                                   M0, 4 * cC, cC, 0);
    k_gemm<<<g64b, 32, 0, stream>>>(f1b, f1, Wf1 + l * 4096, nullptr, f1, M1, cC, cC, 0);
    float* tmp = f1; f1 = f1b; f1b = tmp;   // ping-pong (deterministic per call)
  }

  k_pool_out<<<cB, cC, 0, stream>>>((float*)d_out, f0, mask, Wout, bout);
}